// GATlayer_47777216200928
// MI455X (gfx1250) — compile-verified
//
#include <hip/hip_runtime.h>
#include <hip/hip_bf16.h>

typedef float v2f __attribute__((ext_vector_type(2)));
typedef float v8f __attribute__((ext_vector_type(8)));

#define NFEAT 64
#define EFEAT 32
#define OUTF  64
#define DCAT  160

// ---------------------------------------------------------------------------
// K0: zero output accumulator, segment-max keys (ordered-uint, 0 == key(-inf)
//     lower bound), segment sums.
// ---------------------------------------------------------------------------
__global__ void k_init(float* __restrict__ out, unsigned* __restrict__ segmax,
                       float* __restrict__ segsum, int N) {
    int i = blockIdx.x * blockDim.x + threadIdx.x;
    if (i < N * OUTF) out[i] = 0.0f;
    if (i < N) { segmax[i] = 0u; segsum[i] = 0.0f; }
}

// ---------------------------------------------------------------------------
// K1: wa[0:160] = W_w^T @ a ; wa[160] = W_b . a   (rank-1 collapse of the
// attention projection: logits = leaky_relu(feat_cat . wa + ba))
// ---------------------------------------------------------------------------
__global__ void k_wa(const float* __restrict__ Ww, const float* __restrict__ Wb,
                     const float* __restrict__ a, float* __restrict__ wa) {
    int j = threadIdx.x;
    if (j < DCAT) {
        float s = 0.0f;
        for (int i = 0; i < 2 * OUTF; ++i) s += Ww[i * DCAT + j] * a[i];
        wa[j] = s;
    } else if (j == DCAT) {
        float s = 0.0f;
        for (int i = 0; i < 2 * OUTF; ++i) s += Wb[i] * a[i];
        wa[DCAT] = s;
    }
}

// ---------------------------------------------------------------------------
// K2: per-node attention score halves
// ---------------------------------------------------------------------------
__global__ void k_scores(const float* __restrict__ nf, const float* __restrict__ wa,
                         float* __restrict__ ssrc, float* __restrict__ sdst, int N) {
    int n = blockIdx.x * blockDim.x + threadIdx.x;
    if (n >= N) return;
    const float* row = nf + (size_t)n * NFEAT;
    float a = 0.0f, b = 0.0f;
    #pragma unroll
    for (int j = 0; j < NFEAT; ++j) {
        float v = row[j];
        a += v * wa[j];
        b += v * wa[NFEAT + j];
    }
    ssrc[n] = a;
    sdst[n] = b;
}

// ---------------------------------------------------------------------------
// K3: WMMA node transform: M1 = n_feat @ W1^T, M2 = n_feat @ W2^T
// One wave per 16-node tile. A: 16x4 f32 fragment (v2f/lane), B from
// W_msg_w columns (row-major weight => B[k][n] = W_msg_w[n*160 + koff + k]).
// ---------------------------------------------------------------------------
__global__ void k_node_xform(const float* __restrict__ nf, const float* __restrict__ Wmsg,
                             float* __restrict__ M1, float* __restrict__ M2, int N) {
    int wave = (int)((blockIdx.x * blockDim.x + threadIdx.x) >> 5);
    int lane = threadIdx.x & 31;
    int tiles = N >> 4;
    if (wave >= tiles) return;                 // wave-uniform: EXEC stays all-1s
    int base = wave << 4;
    int row  = lane & 15;
    int kh   = (lane >> 4) << 1;               // 0 or 2: K sub-offset per half-wave

    v8f acc1[4] = {};
    v8f acc2[4] = {};
    const float* arow = nf + (size_t)(base + row) * NFEAT + kh;

    #pragma unroll
    for (int k = 0; k < NFEAT; k += 4) {
        v2f afrag = *(const v2f*)(arow + k);
        #pragma unroll
        for (int t = 0; t < 4; ++t) {
            int n = t * 16 + row;
            v2f b1 = *(const v2f*)(Wmsg + (size_t)n * DCAT + k + kh);          // W1 block
            v2f b2 = *(const v2f*)(Wmsg + (size_t)n * DCAT + NFEAT + k + kh);  // W2 block
            acc1[t] = __builtin_amdgcn_wmma_f32_16x16x4_f32(
                false, afrag, false, b1, (short)0, acc1[t], false, false);
            acc2[t] = __builtin_amdgcn_wmma_f32_16x16x4_f32(
                false, afrag, false, b2, (short)0, acc2[t], false, false);
        }
    }

    int m0 = (lane >> 4) * 8;                  // C/D layout: VGPR i -> M = i or 8+i
    #pragma unroll
    for (int t = 0; t < 4; ++t) {
        #pragma unroll
        for (int i = 0; i < 8; ++i) {
            int m = m0 + i;
            int n = t * 16 + row;
            M1[(size_t)(base + m) * OUTF + n] = acc1[t][i];
            M2[(size_t)(base + m) * OUTF + n] = acc2[t][i];
        }
    }
}

// ---------------------------------------------------------------------------
// K4: per-edge logit + segment max via order-preserving uint atomicMax
// ---------------------------------------------------------------------------
__global__ void k_edge_logit(const float* __restrict__ ef, const int* __restrict__ src,
                             const int* __restrict__ dst, const float* __restrict__ wa,
                             const float* __restrict__ ssrc, const float* __restrict__ sdst,
                             float* __restrict__ logit, unsigned* __restrict__ segmax, int E) {
    int e = blockIdx.x * blockDim.x + threadIdx.x;
    if (e >= E) return;
    const float* row = ef + (size_t)e * EFEAT;
    float s = wa[DCAT];                        // ba
    #pragma unroll
    for (int j = 0; j < EFEAT; ++j) s += row[j] * wa[2 * NFEAT + j];
    int d = dst[e];
    s += ssrc[src[e]] + sdst[d];
    float l = s > 0.0f ? s : 0.01f * s;        // leaky_relu, slope 0.01
    logit[e] = l;
    unsigned u = __float_as_uint(l);
    u = (u & 0x80000000u) ? ~u : (u | 0x80000000u);   // monotonic float->uint
    atomicMax(&segmax[d], u);
}

// ---------------------------------------------------------------------------
// K5: per-edge exp(logit - segmax) + segment sum
// ---------------------------------------------------------------------------
__global__ void k_edge_exp(const int* __restrict__ dst, const unsigned* __restrict__ segmax,
                           float* __restrict__ logit_ex, float* __restrict__ segsum, int E) {
    int e = blockIdx.x * blockDim.x + threadIdx.x;
    if (e >= E) return;
    int d = dst[e];
    unsigned u = segmax[d];
    u = (u & 0x80000000u) ? (u ^ 0x80000000u) : ~u;   // decode
    float mx = __uint_as_float(u);
    float ex = __expf(logit_ex[e] - mx);
    logit_ex[e] = ex;
    atomicAdd(&segsum[d], ex);
}

// ---------------------------------------------------------------------------
// K6: WMMA aggregation. One wave per 16-edge tile:
//   M3 = e_feat_tile[16x32] @ W3^T   (8 K-steps x 4 N-tiles of v_wmma f32)
//   out[dst] += att * (M1[src] + M2[dst] + M3 + bias)
// For a fixed accumulator slot i, lanes 0..15 share one edge row, so the
// M1/M2 gathers and the atomic scatter are coalesced 64B segments.
// ---------------------------------------------------------------------------
__global__ void k_aggregate(const float* __restrict__ ef, const int* __restrict__ src,
                            const int* __restrict__ dst, const float* __restrict__ Wmsg,
                            const float* __restrict__ Wmsgb,
                            const float* __restrict__ M1, const float* __restrict__ M2,
                            const float* __restrict__ ex, const float* __restrict__ segsum,
                            float* __restrict__ out, int E) {
    int wave = (int)((blockIdx.x * blockDim.x + threadIdx.x) >> 5);
    int lane = threadIdx.x & 31;
    int tiles = E >> 4;
    if (wave >= tiles) return;                 // wave-uniform
    int ebase = wave << 4;
    int row   = lane & 15;
    int kh    = (lane >> 4) << 1;

    v8f acc[4] = {};
    const float* arow = ef + (size_t)(ebase + row) * EFEAT + kh;

    #pragma unroll
    for (int k = 0; k < EFEAT; k += 4) {
        v2f afrag = *(const v2f*)(arow + k);
        #pragma unroll
        for (int t = 0; t < 4; ++t) {
            int n = t * 16 + row;
            v2f b = *(const v2f*)(Wmsg + (size_t)n * DCAT + 2 * NFEAT + k + kh); // W3 block
            acc[t] = __builtin_amdgcn_wmma_f32_16x16x4_f32(
                false, afrag, false, b, (short)0, acc[t], false, false);
        }
    }

    int m0 = (lane >> 4) * 8;
    int   s_i[8], d_i[8];
    float att_i[8];
    #pragma unroll
    for (int i = 0; i < 8; ++i) {
        int e = ebase + m0 + i;
        int d = dst[e];
        s_i[i]   = src[e];
        d_i[i]   = d;
        att_i[i] = ex[e] / segsum[d];
    }

    #pragma unroll
    for (int t = 0; t < 4; ++t) {
        int n = t * 16 + row;
        float bn = Wmsgb[n];
        #pragma unroll
        for (int i = 0; i < 8; ++i) {
            float v = att_i[i] * (acc[t][i]
                                  + M1[(size_t)s_i[i] * OUTF + n]
                                  + M2[(size_t)d_i[i] * OUTF + n]
                                  + bn);
            atomicAdd(&out[(size_t)d_i[i] * OUTF + n], v);
        }
    }
}

// ---------------------------------------------------------------------------
// K7: in-place relu on the accumulated output
// ---------------------------------------------------------------------------
__global__ void k_relu(float* __restrict__ out, int n) {
    int i = blockIdx.x * blockDim.x + threadIdx.x;
    if (i < n) out[i] = fmaxf(out[i], 0.0f);
}

extern "C" void kernel_launch(void* const* d_in, const int* in_sizes, int n_in,
                              void* d_out, int out_size, void* d_ws, size_t ws_size,
                              hipStream_t stream) {
    const float* n_feat  = (const float*)d_in[0];
    const float* e_feat  = (const float*)d_in[1];
    const int*   src     = (const int*)  d_in[2];
    const int*   dst     = (const int*)  d_in[3];
    const float* W_msg_w = (const float*)d_in[4];
    const float* W_msg_b = (const float*)d_in[5];
    const float* W_w     = (const float*)d_in[6];
    const float* W_b     = (const float*)d_in[7];
    const float* a_vec   = (const float*)d_in[8];

    const int N = in_sizes[0] / NFEAT;   // 100000 (multiple of 16)
    const int E = in_sizes[2];           // 1600000 (multiple of 16)

    float* out = (float*)d_out;

    // workspace layout (floats)
    float*    ws     = (float*)d_ws;
    float*    wa     = ws;                               // 161 used, reserve 256
    float*    ssrc   = ws + 256;
    float*    sdst   = ssrc + N;
    float*    M1     = sdst + N;
    float*    M2     = M1 + (size_t)N * OUTF;
    float*    exbuf  = M2 + (size_t)N * OUTF;            // logit, then exp
    unsigned* segmax = (unsigned*)(exbuf + E);
    float*    segsum = (float*)(segmax + N);

    const int TB = 256;

    // K0: init accumulators
    k_init<<<(N * OUTF + TB - 1) / TB, TB, 0, stream>>>(out, segmax, segsum, N);
    // K1: rank-1 collapse of attention projection
    k_wa<<<1, 192, 0, stream>>>(W_w, W_b, a_vec, wa);
    // K2: per-node score halves
    k_scores<<<(N + TB - 1) / TB, TB, 0, stream>>>(n_feat, wa, ssrc, sdst, N);
    // K3: WMMA node transforms M1, M2
    {
        int tiles = N >> 4;                       // 6250 waves
        int blocks = (tiles + (TB / 32) - 1) / (TB / 32);
        k_node_xform<<<blocks, TB, 0, stream>>>(n_feat, W_msg_w, M1, M2, N);
    }
    // K4: edge logits + segment max
    k_edge_logit<<<(E + TB - 1) / TB, TB, 0, stream>>>(e_feat, src, dst, wa, ssrc, sdst,
                                                       exbuf, segmax, E);
    // K5: exp + segment sum
    k_edge_exp<<<(E + TB - 1) / TB, TB, 0, stream>>>(dst, segmax, exbuf, segsum, E);
    // K6: WMMA edge messages + weighted scatter
    {
        int tiles = E >> 4;                       // 100000 waves
        int blocks = (tiles + (TB / 32) - 1) / (TB / 32);
        k_aggregate<<<blocks, TB, 0, stream>>>(e_feat, src, dst, W_msg_w, W_msg_b,
                                               M1, M2, exbuf, segsum, out, E);
    }
    // K7: relu
    k_relu<<<(N * OUTF + TB - 1) / TB, TB, 0, stream>>>(out, N * OUTF);
}